// PixelContrastLoss_49632642073101
// MI455X (gfx1250) — compile-verified
//
#include <hip/hip_runtime.h>

typedef __attribute__((ext_vector_type(2))) float v2f;
typedef __attribute__((ext_vector_type(8))) float v8f;

#define N_TOT   8192
#define DIMK    256
#define NTILES  512   // 8192 / 16
#define NWAVES  8

// ---------------------------------------------------------------------------
// Main kernel: one block per 16-row tile of the 8192x8192 Gram matrix.
// Pass A: online row-max + masked negative exp-sum.
// Pass B: recompute logits, accumulate positive log-prob terms.
// ---------------------------------------------------------------------------
__global__ __launch_bounds__(256)
void pcl_main(const float* __restrict__ feats,   // [64,128,256]
              const int*   __restrict__ labels,  // [64]
              float*       __restrict__ part)    // [512] per-block partials
{
    __shared__ float lds_a[16 * DIMK];           // A tile (16 rows of cf)
    __shared__ int   lds_lab[64];
    __shared__ float lds_red[NWAVES][16][2];
    __shared__ float lds_fin[16][2];
    __shared__ float lds_row[16];

    const int tid  = threadIdx.x;
    const int lane = tid & 31;
    const int wv   = tid >> 5;
    const int half = lane >> 4;      // 0: lanes 0-15, 1: lanes 16-31
    const int l15  = lane & 15;
    const int koff = half * 2;       // K-pair offset per WMMA f32 16x16x4 layout
    const int r0   = blockIdx.x * 16;

    if (tid < 64) lds_lab[tid] = labels[tid];
    // Stage A tile: cf[r0+m][k] = feats[((r0+m)&63)*128 + ((r0+m)>>6)][k]
    for (int idx = tid; idx < 16 * DIMK; idx += 256) {
        int m   = idx >> 8;
        int k   = idx & 255;
        int row = r0 + m;
        lds_a[idx] = feats[(((row & 63) * 128 + (row >> 6)) << 8) + k];
    }
    __syncthreads();

    int rowg[8], rlab[8];
#pragma unroll
    for (int v = 0; v < 8; ++v) {                // C/D layout: VGPR v -> row v (+8 for upper half)
        rowg[v] = r0 + v + half * 8;
        rlab[v] = lds_lab[rowg[v] & 63];
    }
    const float* ap = lds_a + l15 * DIMK + koff;

    float rmax[8], rneg[8];
#pragma unroll
    for (int v = 0; v < 8; ++v) { rmax[v] = -3.0e38f; rneg[v] = 0.0f; }

    // ---------------- Pass A ----------------
    for (int nt = wv; nt < NTILES; nt += NWAVES) {
        const int col  = nt * 16 + l15;
        const int clab = lds_lab[col & 63];
        const float* bp = feats + (((col & 63) * 128 + (col >> 6)) << 8) + koff;
        v8f acc = {};
#pragma unroll 4
        for (int kk = 0; kk < DIMK; kk += 4) {
            v2f a = *(const v2f*)(ap + kk);
            v2f b = *(const v2f*)(bp + kk);
            acc = __builtin_amdgcn_wmma_f32_16x16x4_f32(
                false, a, false, b, (short)0, acc, false, false);
        }
#pragma unroll
        for (int v = 0; v < 8; ++v) {
            float l  = acc[v] * 10.0f;                       // 1/TEMPERATURE
            float nm = fmaxf(rmax[v], l);
            float e  = (rlab[v] != clab) ? __expf(l - nm) : 0.0f;
            rneg[v]  = rneg[v] * __expf(rmax[v] - nm) + e;
            rmax[v]  = nm;
        }
    }
    // Butterfly-combine (max, sum) across the 16 lanes of each half
#pragma unroll
    for (int msk = 1; msk <= 8; msk <<= 1) {
#pragma unroll
        for (int v = 0; v < 8; ++v) {
            float mo = __shfl_xor(rmax[v], msk, 32);
            float so = __shfl_xor(rneg[v], msk, 32);
            float nm = fmaxf(rmax[v], mo);
            rneg[v]  = rneg[v] * __expf(rmax[v] - nm) + so * __expf(mo - nm);
            rmax[v]  = nm;
        }
    }
    if (l15 == 0) {
#pragma unroll
        for (int v = 0; v < 8; ++v) {
            lds_red[wv][half * 8 + v][0] = rmax[v];
            lds_red[wv][half * 8 + v][1] = rneg[v];
        }
    }
    __syncthreads();
    if (tid < 16) {
        float m = -3.0e38f, s = 0.0f;
        for (int w = 0; w < NWAVES; ++w) {
            float mw = lds_red[w][tid][0], sw = lds_red[w][tid][1];
            float nm = fmaxf(m, mw);
            s = s * __expf(m - nm) + sw * __expf(mw - nm);
            m = nm;
        }
        lds_fin[tid][0] = m;
        lds_fin[tid][1] = s;
    }
    __syncthreads();

    float fm[8], fneg[8];
#pragma unroll
    for (int v = 0; v < 8; ++v) {
        fm[v]   = lds_fin[half * 8 + v][0];
        fneg[v] = lds_fin[half * 8 + v][1];
    }

    // ---------------- Pass B ----------------
    float psum[8], pcnt[8];
#pragma unroll
    for (int v = 0; v < 8; ++v) { psum[v] = 0.0f; pcnt[v] = 0.0f; }

    for (int nt = wv; nt < NTILES; nt += NWAVES) {
        const int col  = nt * 16 + l15;
        const int clab = lds_lab[col & 63];
        const float* bp = feats + (((col & 63) * 128 + (col >> 6)) << 8) + koff;
        v8f acc = {};
#pragma unroll 4
        for (int kk = 0; kk < DIMK; kk += 4) {
            v2f a = *(const v2f*)(ap + kk);
            v2f b = *(const v2f*)(bp + kk);
            acc = __builtin_amdgcn_wmma_f32_16x16x4_f32(
                false, a, false, b, (short)0, acc, false, false);
        }
#pragma unroll
        for (int v = 0; v < 8; ++v) {
            float l   = acc[v] * 10.0f - fm[v];
            bool pos  = (rlab[v] == clab) & (col != rowg[v]);
            float t   = l - __logf(__expf(l) + fneg[v] + 1.0e-10f);
            psum[v]  += pos ? t : 0.0f;
            pcnt[v]  += pos ? 1.0f : 0.0f;
        }
    }
#pragma unroll
    for (int msk = 1; msk <= 8; msk <<= 1) {
#pragma unroll
        for (int v = 0; v < 8; ++v) {
            psum[v] += __shfl_xor(psum[v], msk, 32);
            pcnt[v] += __shfl_xor(pcnt[v], msk, 32);
        }
    }
    if (l15 == 0) {
#pragma unroll
        for (int v = 0; v < 8; ++v) {
            lds_red[wv][half * 8 + v][0] = psum[v];
            lds_red[wv][half * 8 + v][1] = pcnt[v];
        }
    }
    __syncthreads();
    if (tid < 16) {
        float S = 0.0f, C = 0.0f;
        for (int w = 0; w < NWAVES; ++w) {
            S += lds_red[w][tid][0];
            C += lds_red[w][tid][1];
        }
        lds_row[tid] = S / C;                    // mean_log_prob_pos for this row
    }
    __syncthreads();
    if (tid == 0) {
        float t = 0.0f;
        for (int r = 0; r < 16; ++r) t += lds_row[r];
        part[blockIdx.x] = t;
    }
}

// ---------------------------------------------------------------------------
// Final reduction: 512 partials -> scalar loss (deterministic, no atomics).
// ---------------------------------------------------------------------------
__global__ __launch_bounds__(256)
void pcl_reduce(const float* __restrict__ part, float* __restrict__ out)
{
    __shared__ float s[256];
    float v = part[threadIdx.x] + part[threadIdx.x + 256];
    s[threadIdx.x] = v;
    __syncthreads();
    for (int off = 128; off > 0; off >>= 1) {
        if (threadIdx.x < off) s[threadIdx.x] += s[threadIdx.x + off];
        __syncthreads();
    }
    if (threadIdx.x == 0)
        out[0] = -(0.1f / 0.07f) * s[0] / 8192.0f;   // -(T/BASE_T) * mean
}

extern "C" void kernel_launch(void* const* d_in, const int* in_sizes, int n_in,
                              void* d_out, int out_size, void* d_ws, size_t ws_size,
                              hipStream_t stream) {
    const float* feats  = (const float*)d_in[0];   // [64,128,256] f32
    const int*   labels = (const int*)d_in[1];     // [64] i32
    float*       part   = (float*)d_ws;            // 512 floats of scratch
    float*       out    = (float*)d_out;

    pcl_main<<<NTILES, 256, 0, stream>>>(feats, labels, part);
    pcl_reduce<<<1, 256, 0, stream>>>(part, out);
}